// MATLoss_41180146434453
// MI455X (gfx1250) — compile-verified
//
#include <hip/hip_runtime.h>

// ATLoss for MI455X (gfx1250, wave32). HBM-streaming reduction:
// 268 MB read once -> ~11.5 us roofline at 23.3 TB/s.
// One wave32 per row (C=512 -> 16 elems/lane). CDNA5 async path:
// each wave streams rows into a double-buffered LDS stage with
// GLOBAL_LOAD_ASYNC_TO_LDS_B128 (ASYNCcnt-tracked), overlapping the next
// row's fetch with the current row's transcendental-heavy compute.
// Deterministic double-precision two-stage final reduce (no float atomics).

typedef float v4f __attribute__((ext_vector_type(4)));
typedef int   v4i __attribute__((ext_vector_type(4)));

typedef __attribute__((address_space(1))) v4i* gptr_v4i;   // global
typedef __attribute__((address_space(3))) v4i* lptr_v4i;   // LDS

#define B_ROWS 65536
#define C_COLS 512
#define NEG_BIG 1e30f
#define THREADS 128
#define WAVES_PER_BLOCK (THREADS / 32)              // 4
#define NBLOCKS 4096
#define TOTAL_WAVES (NBLOCKS * WAVES_PER_BLOCK)     // 16384
#define ROWS_PER_WAVE (B_ROWS / TOTAL_WAVES)        // 4
#define VPR (C_COLS / 4)                            // 128 v4f per row

#if __has_builtin(__builtin_amdgcn_global_load_async_to_lds_b128) && \
    __has_builtin(__builtin_amdgcn_s_wait_asynccnt)
#define USE_ASYNC 1
#else
#define USE_ASYNC 0
#endif

__device__ __forceinline__ float log_sigmoid_f(float x) {
    // stable: min(x,0) - log1p(exp(-|x|))
    return fminf(x, 0.0f) - log1pf(__expf(-fabsf(x)));
}

__global__ __launch_bounds__(THREADS) void atloss_stage1(
    const float* __restrict__ logits,
    const float* __restrict__ labels,
    double* __restrict__ partials)
{
    const int lane = threadIdx.x & 31;
    const int wid  = threadIdx.x >> 5;
    const int gw   = blockIdx.x * WAVES_PER_BLOCK + wid;   // global wave id

#if USE_ASYNC
    // Per-wave double buffer: 2 x (2KB logits + 2KB labels) = 8KB/wave,
    // 32KB per 128-thread block.
    __shared__ v4f stage_lg[WAVES_PER_BLOCK][2][VPR];
    __shared__ v4f stage_lb[WAVES_PER_BLOCK][2][VPR];

    auto issue_row = [&](int row, int buf) {
        const v4i* lg = (const v4i*)(logits + (size_t)row * C_COLS);
        const v4i* lb = (const v4i*)(labels + (size_t)row * C_COLS);
        #pragma unroll
        for (int c = 0; c < 4; ++c) {
            __builtin_amdgcn_global_load_async_to_lds_b128(
                (gptr_v4i)(lg + c * 32 + lane),
                (lptr_v4i)(&stage_lg[wid][buf][c * 32 + lane]),
                0, 0);
            __builtin_amdgcn_global_load_async_to_lds_b128(
                (gptr_v4i)(lb + c * 32 + lane),
                (lptr_v4i)(&stage_lb[wid][buf][c * 32 + lane]),
                0, 0);
        }
    };

    issue_row(gw, 0);   // prime the pipeline
#endif

    double acc_term = 0.0;   // sum over positive rows of `term`
    double acc_cnt  = 0.0;   // sum over positive rows of (1 + pos_count)
    double acc_l2   = 0.0;   // sum over rows of loss2

    for (int r = 0; r < ROWS_PER_WAVE; ++r) {
        const int row = gw + r * TOTAL_WAVES;

#if USE_ASYNC
        if (r + 1 < ROWS_PER_WAVE) {
            issue_row(gw + (r + 1) * TOTAL_WAVES, (r + 1) & 1);
            __builtin_amdgcn_s_wait_asynccnt(8);   // current row's 8 ops done
        } else {
            __builtin_amdgcn_s_wait_asynccnt(0);
        }
        __asm__ __volatile__("" ::: "memory");     // pin LDS reads after wait
        const v4f* xrow = &stage_lg[wid][r & 1][0];
        const v4f* yrow = &stage_lb[wid][r & 1][0];
#else
        const v4f* xrow = (const v4f*)(logits + (size_t)row * C_COLS);
        const v4f* yrow = (const v4f*)(labels + (size_t)row * C_COLS);
#endif

        float pos = 0.0f, tsum = 0.0f, zmax = -INFINITY;
        float logit0 = 0.0f;
        float zs[16];

        // Pass 1: accumulate pos/tsum, record z = x - y*1e30, track row max.
        #pragma unroll
        for (int c = 0; c < 4; ++c) {
#if USE_ASYNC
            v4f xv = xrow[c * 32 + lane];
            v4f yv = yrow[c * 32 + lane];
#else
            v4f xv = __builtin_nontemporal_load(&xrow[c * 32 + lane]);
            v4f yv = __builtin_nontemporal_load(&yrow[c * 32 + lane]);
#endif
            if (c == 0 && lane == 0) {      // column 0: threshold class
                logit0 = xv[0];
                yv[0]  = 0.0f;              // labels[:,0] forced to 0
            }
            #pragma unroll
            for (int e = 0; e < 4; ++e) {
                const float x = xv[e], y = yv[e];
                pos  += y;
                tsum += y * log_sigmoid_f(x);
                const float z = x - y * NEG_BIG;
                zs[c * 4 + e] = z;
                zmax = fmaxf(zmax, z);
            }
        }

        // wave32 reductions (all lanes active)
        #pragma unroll
        for (int off = 16; off; off >>= 1) {
            pos  += __shfl_xor(pos,  off, 32);
            tsum += __shfl_xor(tsum, off, 32);
            zmax  = fmaxf(zmax, __shfl_xor(zmax, off, 32));
        }

        // Pass 2 (registers only): sum exp(z - rowmax)
        float es = 0.0f;
        #pragma unroll
        for (int k = 0; k < 16; ++k) es += __expf(zs[k] - zmax);
        #pragma unroll
        for (int off = 16; off; off >>= 1) es += __shfl_xor(es, off, 32);

        if (lane == 0) {
            const float term = log_sigmoid_f(-logit0) + tsum;
            if (pos > 0.0f) {
                acc_term += (double)term;
                acc_cnt  += (double)(1.0f + pos);
            }
            // loss2 = logsumexp(logit2) - logit2[:,0]; logit2[:,0] == logit0
            acc_l2 += (double)(__logf(es) + zmax - logit0);
        }
    }

    __shared__ double st[WAVES_PER_BLOCK];
    __shared__ double sc[WAVES_PER_BLOCK];
    __shared__ double sl[WAVES_PER_BLOCK];
    if (lane == 0) { st[wid] = acc_term; sc[wid] = acc_cnt; sl[wid] = acc_l2; }
    __syncthreads();
    if (threadIdx.x == 0) {
        double t = 0.0, c = 0.0, l = 0.0;
        #pragma unroll
        for (int w = 0; w < WAVES_PER_BLOCK; ++w) { t += st[w]; c += sc[w]; l += sl[w]; }
        partials[(size_t)blockIdx.x * 3 + 0] = t;
        partials[(size_t)blockIdx.x * 3 + 1] = c;
        partials[(size_t)blockIdx.x * 3 + 2] = l;
    }
}

__global__ __launch_bounds__(256) void atloss_stage2(
    const double* __restrict__ partials, float* __restrict__ out)
{
    __shared__ double s0[256];
    __shared__ double s1[256];
    __shared__ double s2[256];
    const int t = threadIdx.x;
    double a0 = 0.0, a1 = 0.0, a2 = 0.0;
    for (int i = t; i < NBLOCKS; i += 256) {   // fixed order -> deterministic
        a0 += partials[(size_t)i * 3 + 0];
        a1 += partials[(size_t)i * 3 + 1];
        a2 += partials[(size_t)i * 3 + 2];
    }
    s0[t] = a0; s1[t] = a1; s2[t] = a2;
    __syncthreads();
    for (int off = 128; off; off >>= 1) {
        if (t < off) { s0[t] += s0[t + off]; s1[t] += s1[t + off]; s2[t] += s2[t + off]; }
        __syncthreads();
    }
    if (t == 0) {
        const double loss1 = -s0[0] / s1[0];
        const double loss2 = s2[0] / (double)B_ROWS;
        out[0] = (float)(loss1 + loss2);
    }
}

extern "C" void kernel_launch(void* const* d_in, const int* in_sizes, int n_in,
                              void* d_out, int out_size, void* d_ws, size_t ws_size,
                              hipStream_t stream) {
    (void)in_sizes; (void)n_in; (void)out_size; (void)ws_size;
    const float* logits = (const float*)d_in[0];
    const float* labels = (const float*)d_in[1];
    float* out          = (float*)d_out;
    double* partials    = (double*)d_ws;   // NBLOCKS*3 doubles = 96 KB

    atloss_stage1<<<NBLOCKS, THREADS, 0, stream>>>(logits, labels, partials);
    atloss_stage2<<<1, 256, 0, stream>>>(partials, out);
}